// TokenMaxSimLoss_73512660239075
// MI455X (gfx1250) — compile-verified
//
#include <hip/hip_runtime.h>
#include <hip/hip_bf16.h>
#include <math.h>

// Problem constants from the reference
#define B_ 256
#define N_ 256
#define D_ 512
#define C_ 1024
#define EPS_ 1e-12f

typedef __attribute__((ext_vector_type(2))) float v2f;
typedef __attribute__((ext_vector_type(8))) float v8f;

// ---------------------------------------------------------------------------
// Kernel A: L2-normalize text features: txt_n[c,:] = txt[c,:] / max(||.||,eps)
// grid = C_, block = 256 (each thread owns 2 of the 512 elements)
// ---------------------------------------------------------------------------
__global__ void norm_text_kernel(const float* __restrict__ txt,
                                 float* __restrict__ txt_n) {
    __shared__ float red[256];
    const int c   = blockIdx.x;
    const int tid = threadIdx.x;
    const float v0 = txt[c * D_ + tid];
    const float v1 = txt[c * D_ + tid + 256];
    red[tid] = v0 * v0 + v1 * v1;
    __syncthreads();
    for (int s = 128; s > 0; s >>= 1) {
        if (tid < s) red[tid] += red[tid + s];
        __syncthreads();
    }
    const float inv = 1.0f / fmaxf(sqrtf(red[0]), EPS_);
    txt_n[c * D_ + tid]       = v0 * inv;
    txt_n[c * D_ + tid + 256] = v1 * inv;
}

// ---------------------------------------------------------------------------
// Kernel B: per-batch fused (sims_gt -> softmax -> weighted token pooling)
// grid = B_, block = 256 threads = 8 wave32s
// ---------------------------------------------------------------------------
__global__ void pool_kernel(const float* __restrict__ tok,     // [B,N,D]
                            const float* __restrict__ txt_n,   // [C,D] normalized
                            const float* __restrict__ log_attn_tau,
                            const int*   __restrict__ pid,     // [B]
                            float* __restrict__ pooled) {      // [B,D]
    __shared__ float g[D_];       // gt text row
    __shared__ float sgt[N_];     // sims_gt for this batch
    __shared__ float invn[N_];    // 1/||tok||
    __shared__ float coef[N_];    // w[n] * invn[n]
    __shared__ float red[256];

    const int b    = blockIdx.x;
    const int tid  = threadIdx.x;
    const int lane = tid & 31;
    const int wave = tid >> 5;
    const int gt   = pid[b];
    const float* tb = tok + (size_t)b * N_ * D_;

    // stage gt text row into LDS
    g[tid]       = txt_n[gt * D_ + tid];
    g[tid + 256] = txt_n[gt * D_ + tid + 256];
    __syncthreads();

    // Stage 1: per-token dot-with-gt and norm, one wave per token, 32 tokens/wave
    for (int t = 0; t < 32; ++t) {
        const int n = wave * 32 + t;
        const float* tp = tb + (size_t)n * D_;
        float adot = 0.0f, anrm = 0.0f;
        #pragma unroll
        for (int j = 0; j < 16; ++j) {
            const int d = lane + 32 * j;      // coalesced 128B per step
            const float v = tp[d];
            adot = fmaf(v, g[d], adot);
            anrm = fmaf(v, v, anrm);
        }
        // wave32 butterfly reduction
        #pragma unroll
        for (int m = 16; m > 0; m >>= 1) {
            adot += __shfl_xor(adot, m, 32);
            anrm += __shfl_xor(anrm, m, 32);
        }
        if (lane == 0) {
            const float inv = 1.0f / fmaxf(sqrtf(anrm), EPS_);
            sgt[n]  = adot * inv;
            invn[n] = inv;
        }
    }
    __syncthreads();

    // Stage 2: softmax over n (N_ == blockDim.x == 256)
    const float attn_tau = fminf(fmaxf(expf(log_attn_tau[0]), 0.01f), 1.0f);
    const float x = sgt[tid] / attn_tau;
    red[tid] = x;
    __syncthreads();
    for (int s = 128; s > 0; s >>= 1) {
        if (tid < s) red[tid] = fmaxf(red[tid], red[tid + s]);
        __syncthreads();
    }
    const float xmax = red[0];
    __syncthreads();
    const float e = expf(x - xmax);
    red[tid] = e;
    __syncthreads();
    for (int s = 128; s > 0; s >>= 1) {
        if (tid < s) red[tid] += red[tid + s];
        __syncthreads();
    }
    const float esum = red[0];
    coef[tid] = (e / esum) * invn[tid];
    __syncthreads();

    // Stage 3: pooled[b,d] = sum_n coef[n] * tok[b,n,d]; thread owns d=tid, tid+256
    float acc0 = 0.0f, acc1 = 0.0f;
    for (int n = 0; n < N_; ++n) {
        const float cw = coef[n];                 // LDS broadcast
        const float* tp = tb + (size_t)n * D_;
        acc0 = fmaf(cw, tp[tid],       acc0);     // coalesced, L2-resident
        acc1 = fmaf(cw, tp[tid + 256], acc1);
    }
    pooled[b * D_ + tid]       = acc0;
    pooled[b * D_ + tid + 256] = acc1;
}

// ---------------------------------------------------------------------------
// Kernel C: logits = pooled[B,D] @ txt_n^T[D,C] via V_WMMA_F32_16X16X4_F32
// One wave per 16x16 output tile. tiles = (B/16)*(C/16) = 1024.
// block = 256 threads (8 waves) -> grid = 128 blocks.
// ISA f32 A 16x4 layout: lanes 0-15 hold rows, VGPR0/1 = K,K+1 (half-wave 0)
// and K+2,K+3 (half-wave 1). B 4x16 mirrors with rows<->cols.
// ---------------------------------------------------------------------------
__global__ void wmma_logits_kernel(const float* __restrict__ pooled,  // [B,D]
                                   const float* __restrict__ txt_n,   // [C,D]
                                   float* __restrict__ logits) {      // [B,C]
    const int tid  = threadIdx.x;
    const int lane = tid & 31;
    const int wave = tid >> 5;
    const int tile = blockIdx.x * 8 + wave;      // 0..1023
    const int m0   = (tile >> 6) << 4;           // tile / 64 * 16  (row block)
    const int c0   = (tile & 63) << 4;           // tile % 64 * 16  (col block)

    const int r   = lane & 15;        // row within half-wave
    const int sel = lane >> 4;        // 0: K,K+1   1: K+2,K+3

    const float* arow = pooled + (size_t)(m0 + r) * D_;
    const float* brow = txt_n  + (size_t)(c0 + r) * D_;

    v8f acc = {};
    for (int k = 0; k < D_; k += 4) {
        const int kk = k + (sel << 1);
        const v2f a = *reinterpret_cast<const v2f*>(arow + kk);
        const v2f b = *reinterpret_cast<const v2f*>(brow + kk);
        acc = __builtin_amdgcn_wmma_f32_16x16x4_f32(
            /*neg_a=*/false, a, /*neg_b=*/false, b,
            /*c_mod=*/(short)0, acc, /*reuse_a=*/false, /*reuse_b=*/false);
    }

    // 16x16 f32 D layout: VGPR v, lanes 0-15 -> M=v, lanes 16-31 -> M=v+8
    const int col = c0 + r;
    #pragma unroll
    for (int v = 0; v < 8; ++v) {
        const int row = m0 + v + (sel << 3);
        logits[(size_t)row * C_ + col] = acc[v];
    }
}

// ---------------------------------------------------------------------------
// Kernel D: per-row cross-entropy: nll[b] = -log_softmax(logits[b]/tau)[gt[b]]
// grid = B_, block = 256 (each thread owns 4 of the 1024 columns)
// ---------------------------------------------------------------------------
__global__ void nll_kernel(const float* __restrict__ logits,
                           const float* __restrict__ log_tau,
                           const int*   __restrict__ pid,
                           float* __restrict__ nll) {
    __shared__ float red[256];
    const int b   = blockIdx.x;
    const int tid = threadIdx.x;
    const float tau = fminf(fmaxf(expf(log_tau[0]), 0.01f), 1.0f);
    const float inv_tau = 1.0f / tau;
    const float* row = logits + (size_t)b * C_;

    float x0 = row[tid]       * inv_tau;
    float x1 = row[tid + 256] * inv_tau;
    float x2 = row[tid + 512] * inv_tau;
    float x3 = row[tid + 768] * inv_tau;

    red[tid] = fmaxf(fmaxf(x0, x1), fmaxf(x2, x3));
    __syncthreads();
    for (int s = 128; s > 0; s >>= 1) {
        if (tid < s) red[tid] = fmaxf(red[tid], red[tid + s]);
        __syncthreads();
    }
    const float m = red[0];
    __syncthreads();
    red[tid] = expf(x0 - m) + expf(x1 - m) + expf(x2 - m) + expf(x3 - m);
    __syncthreads();
    for (int s = 128; s > 0; s >>= 1) {
        if (tid < s) red[tid] += red[tid + s];
        __syncthreads();
    }
    if (tid == 0) {
        const float lse = m + logf(red[0]);
        const float xg  = row[pid[b]] * inv_tau;
        nll[b] = lse - xg;
    }
}

// ---------------------------------------------------------------------------
// Kernel E: mean over B -> d_out[0]
// ---------------------------------------------------------------------------
__global__ void mean_kernel(const float* __restrict__ nll, float* __restrict__ out) {
    __shared__ float red[256];
    const int tid = threadIdx.x;
    red[tid] = nll[tid];
    __syncthreads();
    for (int s = 128; s > 0; s >>= 1) {
        if (tid < s) red[tid] += red[tid + s];
        __syncthreads();
    }
    if (tid == 0) out[0] = red[0] * (1.0f / (float)B_);
}

// ---------------------------------------------------------------------------
extern "C" void kernel_launch(void* const* d_in, const int* in_sizes, int n_in,
                              void* d_out, int out_size, void* d_ws, size_t ws_size,
                              hipStream_t stream) {
    (void)in_sizes; (void)n_in; (void)out_size; (void)ws_size;
    const float* tok          = (const float*)d_in[0];  // [B,N,D]
    const float* txt          = (const float*)d_in[1];  // [C,D]
    const float* log_tau      = (const float*)d_in[2];  // [1]
    const float* log_attn_tau = (const float*)d_in[3];  // [1]
    const int*   pid          = (const int*)d_in[4];    // [B]
    float* out = (float*)d_out;

    // workspace layout (floats)
    float* txt_n  = (float*)d_ws;                       // C*D   = 524288
    float* pooled = txt_n  + (size_t)C_ * D_;           // B*D   = 131072
    float* logits = pooled + (size_t)B_ * D_;           // B*C   = 262144
    float* nll    = logits + (size_t)B_ * C_;           // B     = 256

    norm_text_kernel<<<C_, 256, 0, stream>>>(txt, txt_n);
    pool_kernel<<<B_, 256, 0, stream>>>(tok, txt_n, log_attn_tau, pid, pooled);
    wmma_logits_kernel<<<(B_ / 16) * (C_ / 16) / 8, 256, 0, stream>>>(pooled, txt_n, logits);
    nll_kernel<<<B_, 256, 0, stream>>>(logits, log_tau, pid, nll);
    mean_kernel<<<1, 256, 0, stream>>>(nll, out);
}